// SpikeActivation_39453569581341
// MI455X (gfx1250) — compile-verified
//
#include <hip/hip_runtime.h>

// Elementwise 10-step leaky integrate-and-fire with spike feedback.
// Memory-bound streaming kernel: 512 MB once-touched traffic (> 192 MB L2),
// so use non-temporal B128 loads/stores; per-element recurrence kept to
// ~6 VALU ops/step with bit-faithful rounding (no fma contraction).

typedef float v4f __attribute__((ext_vector_type(4)));

__device__ __forceinline__ float lif10(float x) {
    // Step 0: mem = 0.99*0 + x - 0 = x
    float mem = x;
    bool  sp  = mem > 1.0f;            // spk_0 = (mem - 1 > 0)  ==  (mem > 1)
    float s   = sp ? 1.0f : 0.0f;
    float acc = s;
#pragma unroll
    for (int t = 1; t < 10; ++t) {
        // reset_t = (mem_{t-1} > 1) == spk_{t-1} == s ; inp_t = spk_{t-1} == s
        // mem = (BETA*mem + inp) - reset*THRESHOLD, each op rounded separately
        mem = __fsub_rn(__fadd_rn(__fmul_rn(0.99f, mem), s), s);
        sp  = mem > 1.0f;
        s   = sp ? 1.0f : 0.0f;
        acc = __fadd_rn(acc, s);
    }
    // mean over 10 steps: correctly-rounded acc/10 equals acc*0.1f for all
    // reachable acc in {0..10} except acc==9 (9*0.1f rounds 1 ulp high).
    float r = __fmul_rn(acc, 0.1f);
    r = (acc == 9.0f) ? 0x1.ccccccp-1f : r;   // 0x3F666666 = RN(9/10)
    return r;
}

__global__ __launch_bounds__(256) void snn_lif10_kernel(
    const v4f* __restrict__ xv, v4f* __restrict__ ov, int n4,
    const float* __restrict__ x, float* __restrict__ out, int n)
{
    const int tid  = threadIdx.x;
    const int base = blockIdx.x * (blockDim.x * 2);
    const int i0   = base + tid;
    const int i1   = i0 + blockDim.x;

    if (i1 < n4) {
        v4f a = __builtin_nontemporal_load(xv + i0);
        v4f b = __builtin_nontemporal_load(xv + i1);
        v4f ra, rb;
        ra[0] = lif10(a[0]); ra[1] = lif10(a[1]);
        ra[2] = lif10(a[2]); ra[3] = lif10(a[3]);
        rb[0] = lif10(b[0]); rb[1] = lif10(b[1]);
        rb[2] = lif10(b[2]); rb[3] = lif10(b[3]);
        __builtin_nontemporal_store(ra, ov + i0);
        __builtin_nontemporal_store(rb, ov + i1);
    } else if (i0 < n4) {
        v4f a = __builtin_nontemporal_load(xv + i0);
        v4f ra;
        ra[0] = lif10(a[0]); ra[1] = lif10(a[1]);
        ra[2] = lif10(a[2]); ra[3] = lif10(a[3]);
        __builtin_nontemporal_store(ra, ov + i0);
    }

    // Scalar tail for n % 4 (robustness; n = 64M is divisible by 4).
    const int rem = n & 3;
    if (blockIdx.x == 0 && tid < rem) {
        const int idx = n4 * 4 + tid;
        out[idx] = lif10(x[idx]);
    }
}

extern "C" void kernel_launch(void* const* d_in, const int* in_sizes, int n_in,
                              void* d_out, int out_size, void* d_ws, size_t ws_size,
                              hipStream_t stream) {
    const float* x   = (const float*)d_in[0];
    float*       out = (float*)d_out;
    const int n  = in_sizes[0];
    const int n4 = n >> 2;

    const int BLOCK     = 256;          // 8 wave32 waves per block
    const int PER_BLOCK = BLOCK * 2;    // two float4 per thread
    int grid = (n4 + PER_BLOCK - 1) / PER_BLOCK;
    if (grid < 1) grid = 1;

    snn_lif10_kernel<<<grid, BLOCK, 0, stream>>>(
        (const v4f*)x, (v4f*)out, n4, x, out, n);
}